// UltraQuery_24507083391242
// MI455X (gfx1250) — compile-verified
//
#include <hip/hip_runtime.h>
#include <hip/hip_bf16.h>
#include <math.h>

typedef __attribute__((ext_vector_type(2))) float v2f;
typedef __attribute__((ext_vector_type(8))) float v8f;

#define EPS_F 1e-10f
#define D_DIM 64

// ---------------------------------------------------------------------------
// Kernel 1: coefficient table via V_WMMA_F32_16X16X4_F32.
//   coef[m][r] = sum_d rel_emb[rsel[m]][d] * w_out[d] * rel_proj[r][d]
//   m = leg*16 + b  (B==16), rsel[m] = r_index[b*2 + leg]
// One wave (32 lanes) computes one 16x16 tile of coef. Grid: (ceil(R/16), 2).
// A-frag (16x4 f32, 2 VGPR): lanes 0-15 -> K=0,1 ; lanes 16-31 -> K=2,3 (row = lane&15)
// B-frag (4x16 f32, 2 VGPR): mirrored (col = lane&15)
// D (16x16 f32, 8 VGPR): VGPR j -> row (lane<16 ? j : j+8), col = lane&15
// Out-of-range relation columns are handled branchlessly: clamp the row
// pointer and multiply the fragment by a 0/1 mask (keeps EXEC uniform and
// the WMMA loop free of saveexec regions).
// ---------------------------------------------------------------------------
__global__ void coef_wmma_kernel(const int* __restrict__ r_index,
                                 const float* __restrict__ rel_emb,
                                 const float* __restrict__ rel_proj,
                                 const float* __restrict__ w_out,
                                 float* __restrict__ coef,
                                 int R, int rstride) {
  const int nt   = blockIdx.x;      // relation tile
  const int mt   = blockIdx.y;      // leg (0/1)
  const int lane = threadIdx.x;     // 0..31, full wave, EXEC all-ones
  const int half = lane >> 4;
  const int lp   = lane & 15;

  const int rowSel = r_index[lp * 2 + mt];        // relation id for query (b=lp, leg=mt)
  const v2f* __restrict__ A2 = (const v2f*)(rel_emb + (long)rowSel * D_DIM);
  const v2f* __restrict__ W2 = (const v2f*)w_out;

  const int   n    = nt * 16 + lp;                // output relation column
  const float mask = (n < R) ? 1.f : 0.f;
  const v2f*  __restrict__ B2 =
      (const v2f*)(rel_proj + (long)(n < R ? n : 0) * D_DIM);

  v8f c = {0.f, 0.f, 0.f, 0.f, 0.f, 0.f, 0.f, 0.f};
  #pragma unroll
  for (int kk = 0; kk < 16; ++kk) {               // 16 steps of K=4 cover D=64
    const int p = kk * 2 + half;                  // packed float2 index (d0/2)
    v2f a = A2[p] * W2[p];                        // fold w_out into A
    v2f b = B2[p];
    b.x *= mask;
    b.y *= mask;
    c = __builtin_amdgcn_wmma_f32_16x16x4_f32(false, a, false, b,
                                              (short)0, c, false, false);
  }

  const int col   = nt * 16 + lp;                 // < rstride (padded), always safe
  const int mbase = mt * 16 + half * 8;
  #pragma unroll
  for (int j = 0; j < 8; ++j) {
    coef[(mbase + j) * rstride + col] = c[j];
  }
}

// ---------------------------------------------------------------------------
// Kernel 0: zero the aggregation buffer (workspace is poisoned, atomics need 0)
// ---------------------------------------------------------------------------
__global__ void zero_kernel(float* __restrict__ p, int n) {
  const int i = blockIdx.x * blockDim.x + threadIdx.x;
  if (i < n) p[i] = 0.f;
}

// ---------------------------------------------------------------------------
// Kernel 2: edge scan. One thread per edge; compare src against all 32 query
// entities (LDS-cached); on match scatter coef[bk][etype] into agg[bk][dst].
// Matches are rare (~E/N per query) so divergence/atomics are negligible.
// ---------------------------------------------------------------------------
__global__ void edge_scan_kernel(const int* __restrict__ e_index,
                                 const int* __restrict__ src,
                                 const int* __restrict__ dst,
                                 const int* __restrict__ etype,
                                 const float* __restrict__ coef,
                                 float* __restrict__ agg,
                                 int E, int N, int rstride) {
  __shared__ int ent[32];
  if (threadIdx.x < 32) {
    const int bk = threadIdx.x;                   // bk = leg*16 + b
    ent[bk] = e_index[(bk & 15) * 2 + (bk >> 4)];
  }
  __syncthreads();

  const int e = blockIdx.x * blockDim.x + threadIdx.x;
  if (e >= E) return;
  const int s = src[e];

  int et = -1, d = 0;
  #pragma unroll 4
  for (int bk = 0; bk < 32; ++bk) {
    if (s == ent[bk]) {
      if (et < 0) { et = etype[e]; d = dst[e]; }  // load dst/type only on match
      atomicAdd(&agg[(long)bk * N + d], coef[bk * rstride + et]);
    }
  }
}

// ---------------------------------------------------------------------------
// Kernel 3: epilogue. t1*t2 product logic + final logit transform.
// agg layout: leg0 at rows [0..B), leg1 at rows [B..2B). out[b*N + n].
// ---------------------------------------------------------------------------
__global__ void final_kernel(const float* __restrict__ agg,
                             float* __restrict__ out, int B, int N) {
  const int i = blockIdx.x * blockDim.x + threadIdx.x;
  if (i >= B * N) return;
  const int b = i / N;
  const int n = i - b * N;
  const float a1 = agg[(long)b * N + n];
  const float a2 = agg[(long)(B + b) * N + n];
  const float t1 = 1.f / (1.f + __expf(-a1));
  const float t2 = 1.f / (1.f + __expf(-a2));
  const float t  = t1 * t2;
  out[i] = __logf((t + EPS_F) / (1.f - t + EPS_F));
}

// ---------------------------------------------------------------------------
extern "C" void kernel_launch(void* const* d_in, const int* in_sizes, int n_in,
                              void* d_out, int out_size, void* d_ws, size_t ws_size,
                              hipStream_t stream) {
  const int*   e_index    = (const int*)d_in[0];   // [B,2]
  const int*   r_index    = (const int*)d_in[1];   // [B,2]
  const int*   edge_index = (const int*)d_in[2];   // [2,E]
  const int*   edge_type  = (const int*)d_in[3];   // [E]
  /* d_in[4] = num_nodes device scalar: unused, N derived from out_size */
  const float* rel_emb    = (const float*)d_in[5]; // [R,64]
  const float* rel_proj   = (const float*)d_in[6]; // [R,64]
  const float* w_out      = (const float*)d_in[7]; // [64]

  const int B = in_sizes[0] / 2;                   // 16
  const int E = in_sizes[3];                       // 300000
  const int R = in_sizes[5] / D_DIM;               // 200
  const int N = out_size / B;                      // 30000

  const int ntiles  = (R + 15) / 16;               // 13
  const int rstride = ntiles * 16;                 // 208 (padded coef stride)

  const int* src = edge_index;
  const int* dst = edge_index + E;

  // Workspace layout: [coef: 32*rstride f32][align 256][agg: 2*B*N f32]
  float* coef = (float*)d_ws;
  size_t aggOff = (size_t)(2 * B) * rstride * sizeof(float);
  aggOff = (aggOff + 255) & ~(size_t)255;
  float* agg = (float*)((char*)d_ws + aggOff);

  const int aggN = 2 * B * N;
  zero_kernel<<<(aggN + 255) / 256, 256, 0, stream>>>(agg, aggN);

  coef_wmma_kernel<<<dim3(ntiles, 2), 32, 0, stream>>>(
      r_index, rel_emb, rel_proj, w_out, coef, R, rstride);

  edge_scan_kernel<<<(E + 255) / 256, 256, 0, stream>>>(
      e_index, src, dst, edge_type, coef, agg, E, N, rstride);

  final_kernel<<<(B * N + 255) / 256, 256, 0, stream>>>(
      agg, (float*)d_out, B, N);
}